// Describe_1915555414391
// MI455X (gfx1250) — compile-verified
//
#include <hip/hip_runtime.h>

#define BSZ 128
#define CC 2048
#define HW 196
#define ND 32
#define NA 1845

typedef __attribute__((ext_vector_type(2))) float v2f;
typedef __attribute__((ext_vector_type(8))) float v8f;

// ---------------- Stage 1: attended[b,c] = mean_hw(mask[b,hw]*feat[b,c,hw]) ---
// One wave32 per (b,c) row; 196 contiguous floats per row, coalesced strided
// lane loads, shuffle tree reduction. Pure HBM stream of `features` (205 MB).
__global__ void __launch_bounds__(256) attend_kernel(
    const float* __restrict__ mask,
    const float* __restrict__ feat,
    float* __restrict__ attended) {
  const int wave = (blockIdx.x * blockDim.x + threadIdx.x) >> 5; // == b*CC + c
  const int lane = threadIdx.x & 31;
  if (wave >= BSZ * CC) return;
  const int b = wave >> 11;  // /2048
  const float* __restrict__ frow = feat + (size_t)wave * HW;
  const float* __restrict__ mrow = mask + (size_t)b * HW;
  float s = 0.f;
  for (int hw = lane; hw < HW; hw += 32) s += mrow[hw] * frow[hw];
  for (int off = 16; off > 0; off >>= 1) s += __shfl_down(s, off, 32);
  if (lane == 0) attended[wave] = s * (1.0f / (float)HW);
}

// ---------------- Stage 2: group samples by descriptor ----------------------
// counts[d] = #samples with instance d; lists[d*128 + slot] = sample id.
// Slot order from atomicAdd may vary, but each sample's result/output address
// is independent of its slot -> output is deterministic.
__global__ void group_kernel(const int* __restrict__ instance,
                             int* __restrict__ counts,
                             int* __restrict__ lists) {
  const int t = threadIdx.x;
  if (t < ND) counts[t] = 0;
  __syncthreads();
  if (t < BSZ) {
    const int d = instance[t];
    const int slot = atomicAdd(&counts[d], 1);
    lists[d * BSZ + slot] = t;
  }
}

// ---------------- Stage 3: grouped GEMM with V_WMMA_F32_16X16X4_F32 ---------
// One wave per 16x16 output tile of (samples-of-descriptor d) x (answers).
// A (16x4 f32): lanes 0-15 = M rows, VGPR0=K(2*half), VGPR1=K(2*half+1)
// B (4x16 f32): lanes 0-15 = N cols, VGPR0=K(2*half), VGPR1=K(2*half+1)
// -> each lane loads one contiguous float2 from its row per K-step of 4.
// D (16x16 f32): VGPR r -> M = r + 8*half, N = lane&15.
// Two independent accumulators break the WMMA->WMMA C-chain; unroll 4 gives
// 8 wmma + 16 b64 loads per loop iteration to keep the HBM stream saturated.
__global__ void __launch_bounds__(32) wmma_gemm_kernel(
    const float* __restrict__ attended,
    const float* __restrict__ Wt,     // [ND][NA][CC]
    const float* __restrict__ bias,   // [ND][NA]
    const int* __restrict__ counts,
    const int* __restrict__ lists,
    float* __restrict__ out) {        // [BSZ][NA]
  const int d = blockIdx.z;
  const int n_d = counts[d];                 // wave-uniform
  const int mbase = blockIdx.y * 16;
  if (mbase >= n_d) return;                  // uniform exit: EXEC stays full
  const int nbase = blockIdx.x * 16;
  const int rows = min(16, n_d - mbase);

  const int lane = threadIdx.x;
  const int half = lane >> 4;                // 0 or 1
  const int mn = lane & 15;

  // A-row gather (clamp padded rows to a valid sample; results masked on store)
  const int mclamp = (mn < rows) ? mn : (rows - 1);
  const int sidxA = lists[d * BSZ + mbase + mclamp];
  const float* __restrict__ arow = attended + (size_t)sidxA * CC;

  // B-row: column nbase+mn of this descriptor's [NA x CC] slab (clamped)
  int ncol = nbase + mn;
  if (ncol >= NA) ncol = NA - 1;
  const float* __restrict__ brow = Wt + ((size_t)d * NA + ncol) * CC;

  const int koff = 2 * half;
  v8f acc0 = {};
  v8f acc1 = {};
#pragma unroll 4
  for (int k0 = 0; k0 < CC; k0 += 8) {
    v2f a0 = *(const v2f*)(arow + k0 + koff);
    v2f b0 = *(const v2f*)(brow + k0 + koff);
    v2f a1 = *(const v2f*)(arow + k0 + 4 + koff);
    v2f b1 = *(const v2f*)(brow + k0 + 4 + koff);
    // 8 args: (neg_a, A, neg_b, B, c_mod, C, reuse_a, reuse_b)
    acc0 = __builtin_amdgcn_wmma_f32_16x16x4_f32(
        false, a0, false, b0, (short)0, acc0, false, false);
    acc1 = __builtin_amdgcn_wmma_f32_16x16x4_f32(
        false, a1, false, b1, (short)0, acc1, false, false);
  }
  v8f acc = acc0 + acc1;

  const int nout = nbase + mn;
  if (nout < NA) {
    const float bv = bias[(size_t)d * NA + nout];
#pragma unroll
    for (int r = 0; r < 8; ++r) {
      const int mo = r + 8 * half;
      if (mo < rows) {
        const int s = lists[d * BSZ + mbase + mo];
        out[(size_t)s * NA + nout] = acc[r] + bv;
      }
    }
  }
}

extern "C" void kernel_launch(void* const* d_in, const int* in_sizes, int n_in,
                              void* d_out, int out_size, void* d_ws, size_t ws_size,
                              hipStream_t stream) {
  const float* mask     = (const float*)d_in[0];   // [128,1,14,14]
  const float* feat     = (const float*)d_in[1];   // [128,2048,14,14]
  const int*   instance = (const int*)d_in[2];     // [128]
  const float* Wt       = (const float*)d_in[3];   // [32,1845,2048]
  const float* bias     = (const float*)d_in[4];   // [32,1845]
  float* out = (float*)d_out;                      // [128,1845]

  // workspace: attended (1 MB f32) | counts (32 int) | lists (32*128 int)
  float* attended = (float*)d_ws;
  int* counts = (int*)((char*)d_ws + (size_t)BSZ * CC * sizeof(float));
  int* lists  = counts + ND;

  const int totalWaves = BSZ * CC;                 // 262144 rows
  const int threads = 256;
  const int blocks = (totalWaves * 32) / threads;  // 32768
  attend_kernel<<<blocks, threads, 0, stream>>>(mask, feat, attended);

  group_kernel<<<1, 128, 0, stream>>>(instance, counts, lists);

  dim3 grid((NA + 15) / 16, (BSZ + 15) / 16, ND);  // 116 x 8 x 32
  wmma_gemm_kernel<<<grid, 32, 0, stream>>>(attended, Wt, bias, counts, lists, out);
}